// PanguProMoESparseMoeBlock_56040733278529
// MI455X (gfx1250) — compile-verified
//
#include <hip/hip_runtime.h>
#include <hip/hip_bf16.h>

#define T_ 256
#define H_ 1024
#define I_ 512
#define E_ 64
#define TOPK_ 8
#define EPG_ 8
#define AP 516  // LDS row pitch (floats): 516 % 64 == 4 -> conflict-free column reads

typedef __attribute__((ext_vector_type(2))) float v2f;
typedef __attribute__((ext_vector_type(8))) float v8f;
typedef int v4i_ __attribute__((vector_size(4 * sizeof(int))));  // matches builtin param type

#if defined(__gfx1250__) && \
    __has_builtin(__builtin_amdgcn_global_load_async_to_lds_b128) && \
    __has_builtin(__builtin_amdgcn_s_wait_asynccnt)
#define USE_ASYNC_LDS 1
#else
#define USE_ASYNC_LDS 0
#endif

static __device__ inline v8f v8zero() {
    v8f z;
    for (int i = 0; i < 8; ++i) z[i] = 0.f;
    return z;
}

// ---------------- kernel 0: reset per-expert counters ----------------
__global__ __launch_bounds__(64) void zero_counts(int* cnt) {
    cnt[threadIdx.x] = 0;
}

// ---------------- kernel 1: router (logits, softmax, grouped top-1) ----------------
__global__ __launch_bounds__(64) void router_kernel(
    const float* __restrict__ hs, const float* __restrict__ gw,
    const float* __restrict__ rs, int* __restrict__ cnt,
    int* __restrict__ lidx, float* __restrict__ lwgt)
{
    __shared__ float sprob[E_];
    const int t = blockIdx.x;
    const int e = threadIdx.x;

    const float4* hrow = (const float4*)(hs + (size_t)t * H_);
    const float4* grow = (const float4*)(gw + (size_t)e * H_);
    float acc = 0.f;
    for (int j = 0; j < H_ / 4; ++j) {
        float4 a = hrow[j], b = grow[j];
        acc += a.x * b.x + a.y * b.y + a.z * b.z + a.w * b.w;
    }
    sprob[e] = acc;
    __syncthreads();
    float mx = -3.4e38f;
    for (int j = 0; j < E_; ++j) mx = fmaxf(mx, sprob[j]);
    float p = expf(acc - mx);
    __syncthreads();
    sprob[e] = p;
    __syncthreads();
    float sum = 0.f;
    for (int j = 0; j < E_; ++j) sum += sprob[j];
    float prob = p / sum;
    __syncthreads();
    sprob[e] = prob;
    __syncthreads();

    if (e < TOPK_) {               // one thread per group of 8 experts
        const int base = e * EPG_;
        float best = sprob[base];
        int bi = 0;
        for (int j = 1; j < EPG_; ++j) {
            float v = sprob[base + j];
            if (v > best) { best = v; bi = j; }   // strict > keeps first max (jnp.argmax)
        }
        const int id = base + bi;
        const float w = best * rs[id];
        const int pos = atomicAdd(&cnt[id], 1);
        lidx[id * T_ + pos] = t * TOPK_ + e;      // encodes (token, slot)
        lwgt[id * T_ + pos] = w;
    }
}

// ---------------- kernel 2: per-expert fused gate_up -> silu -> down ----------------
__global__ __launch_bounds__(256) void expert_kernel(
    const float* __restrict__ hs, const float* __restrict__ wgu,
    const float* __restrict__ wdn, const int* __restrict__ cnt,
    const int* __restrict__ lidx, const float* __restrict__ lwgt,
    float* __restrict__ dscr)
{
    __shared__ float smem[16 * AP];   // hidden tile (16x512 chunk), reused as act tile (16x512)
    __shared__ int   entLds[16];
    __shared__ int   tokLds[16];
    __shared__ float wLds[16];

    const int e     = blockIdx.y;
    const int start = blockIdx.x * 16;
    const int n     = cnt[e];
    if (start >= n) return;

    const int tid  = threadIdx.x;
    const int lane = tid & 31;
    const int wv   = tid >> 5;      // wave id 0..7
    const int hi   = lane >> 4;     // half-wave: selects K offset {0,1} vs {2,3}
    const int m    = lane & 15;     // A row / B,C,D column

    if (tid < 16) {
        const int gi = start + tid;
        if (gi < n) {
            const int ent = lidx[e * T_ + gi];
            entLds[tid] = ent;
            tokLds[tid] = ent >> 3;
            wLds[tid]   = lwgt[e * T_ + gi];
        } else {
            entLds[tid] = -1;
            tokLds[tid] = 0;
            wLds[tid]   = 0.f;
        }
    }
    __syncthreads();

    const float* wg = wgu + (size_t)e * (2 * I_) * H_;
    const float* wd = wdn + (size_t)e * H_ * I_;

    // GEMM1: gu[t][o] = sum_h hs[t][h] * wg[o][h];  M=16 tokens, N=2I, K=H (split in 2 chunks of 512)
    v8f accg[4], accu[4];
    for (int p = 0; p < 4; ++p) { accg[p] = v8zero(); accu[p] = v8zero(); }

    const int rowA = tid >> 4;          // 16 threads per row
    const int colA = (tid & 15) * 4;

    for (int kc = 0; kc < 2; ++kc) {
        __syncthreads();                 // previous chunk's readers done
        const float* src = hs + (size_t)tokLds[rowA] * H_ + kc * 512;
        for (int j = 0; j < 8; ++j) {
            const int c = colA + j * 64;
#if USE_ASYNC_LDS
            // cache -> LDS without a VGPR round trip; tracked by ASYNCcnt
            __builtin_amdgcn_global_load_async_to_lds_b128(
                (__attribute__((address_space(1))) v4i_*)(src + c),
                (__attribute__((address_space(3))) v4i_*)&smem[rowA * AP + c],
                /*offset=*/0, /*cpol=*/0);
#else
            *(float4*)&smem[rowA * AP + c] = *(const float4*)(src + c);
#endif
        }
#if USE_ASYNC_LDS
        __builtin_amdgcn_s_wait_asynccnt(0);   // this wave's LDS writes landed
#endif
        __syncthreads();

        for (int p = 0; p < 4; ++p) {
            const int sg = (wv * 4 + p) * 16;                 // gate slab base (0..496)
            const float* a_base  = &smem[m * AP + 2 * hi];
            const float* bg_base = wg + (size_t)(sg + m) * H_ + kc * 512 + 2 * hi;
            const float* bu_base = wg + (size_t)(512 + sg + m) * H_ + kc * 512 + 2 * hi;
            if (p < 3) {   // pull next slab's weight rows toward the WGP a slab early
                __builtin_prefetch(wg + (size_t)(sg + 16 + m) * H_ + kc * 512, 0, 1);
                __builtin_prefetch(wg + (size_t)(512 + sg + 16 + m) * H_ + kc * 512, 0, 1);
            }
            v8f cg = accg[p], cu = accu[p];
            for (int kb = 0; kb < 512; kb += 4) {
                v2f a  = *(const v2f*)(a_base + kb);
                v2f bg = *(const v2f*)(bg_base + kb);
                v2f bu = *(const v2f*)(bu_base + kb);
                cg = __builtin_amdgcn_wmma_f32_16x16x4_f32(false, a, false, bg, (short)0, cg, false, false);
                cu = __builtin_amdgcn_wmma_f32_16x16x4_f32(false, a, false, bu, (short)0, cu, false, false);
            }
            accg[p] = cg; accu[p] = cu;
        }
    }
    __syncthreads();

    // SiLU(gate) * up, written as the GEMM2 A-tile (16 x 512) into the same LDS
    for (int p = 0; p < 4; ++p) {
        const int sg = (wv * 4 + p) * 16;
        for (int r = 0; r < 8; ++r) {
            const int tl = r + 8 * hi;                        // D row -> local token
            const float g = accg[p][r];
            const float u = accu[p][r];
            smem[tl * AP + sg + m] = (g / (1.f + expf(-g))) * u;
        }
    }
    __syncthreads();

    // GEMM2: down[t][h] = sum_i act[t][i] * wd[h][i];  M=16, N=H (64 slabs), K=I
    for (int p = 0; p < 8; ++p) {
        const int hb = (wv * 8 + p) * 16;                     // h slab base (0..1008)
        v8f acc = v8zero();
        const float* a_base = &smem[m * AP + 2 * hi];
        const float* b_base = wd + (size_t)(hb + m) * I_ + 2 * hi;
        if (p < 7)
            __builtin_prefetch(wd + (size_t)(hb + 16 + m) * I_, 0, 1);
        for (int kb = 0; kb < I_; kb += 4) {
            v2f a = *(const v2f*)(a_base + kb);
            v2f b = *(const v2f*)(b_base + kb);
            acc = __builtin_amdgcn_wmma_f32_16x16x4_f32(false, a, false, b, (short)0, acc, false, false);
        }
        for (int r = 0; r < 8; ++r) {
            const int tl = r + 8 * hi;
            const int ent = entLds[tl];
            if (ent >= 0)
                dscr[(size_t)ent * H_ + hb + m] = wLds[tl] * acc[r];
        }
    }
}

// ---------------- kernel 3: combine the 8 slots per token ----------------
__global__ __launch_bounds__(256) void reduce_kernel(
    const float* __restrict__ dscr, float* __restrict__ out)
{
    const int idx = blockIdx.x * 256 + threadIdx.x;  // over T*H
    const int t = idx >> 10;
    const int h = idx & (H_ - 1);
    float s = 0.f;
    for (int k = 0; k < TOPK_; ++k)
        s += dscr[((size_t)t * TOPK_ + k) * H_ + h];
    out[idx] = s;
}

extern "C" void kernel_launch(void* const* d_in, const int* in_sizes, int n_in,
                              void* d_out, int out_size, void* d_ws, size_t ws_size,
                              hipStream_t stream) {
    const float* hs  = (const float*)d_in[0];   // (T,H)
    const float* gw  = (const float*)d_in[1];   // (E,H)
    const float* rs  = (const float*)d_in[2];   // (E,)
    const float* wgu = (const float*)d_in[3];   // (E,2I,H)
    const float* wdn = (const float*)d_in[4];   // (E,H,I)
    float* out = (float*)d_out;                 // (T,H)

    char* ws = (char*)d_ws;
    int*   cnt  = (int*)(ws + 0);                       //  64 * 4
    int*   lidx = (int*)(ws + 256);                     //  64 * 256 * 4
    float* lwgt = (float*)(ws + 256 + 65536);           //  64 * 256 * 4
    float* dscr = (float*)(ws + 256 + 65536 + 65536);   //  T * 8 * H * 4 = 8 MB

    zero_counts<<<1, 64, 0, stream>>>(cnt);
    router_kernel<<<T_, 64, 0, stream>>>(hs, gw, rs, cnt, lidx, lwgt);
    expert_kernel<<<dim3(16, E_), 256, 0, stream>>>(hs, wgu, wdn, cnt, lidx, lwgt, dscr);
    reduce_kernel<<<(T_ * H_) / 256, 256, 0, stream>>>(dscr, out);
}